// MultiBoxLoss_22883585753337
// MI455X (gfx1250) — compile-verified
//
#include <hip/hip_runtime.h>

#define NOBJ   32
#define NCLS   4
#define SORT_M 16384
#define TPB    256
#define THRESH 0.5f
#define BMAX   256   // max batch rows supported by finalize staging

#if __has_builtin(__builtin_amdgcn_global_load_async_to_lds_b128) && __has_builtin(__builtin_amdgcn_s_wait_asynccnt)
#define USE_ASYNC_LDS 1
#endif
#if __has_builtin(__builtin_amdgcn_wmma_f32_16x16x4_f32)
#define USE_WMMA_RED 1
#endif

typedef float v2f __attribute__((ext_vector_type(2)));
typedef float v8f __attribute__((ext_vector_type(8)));
typedef int   v4i_b128 __attribute__((vector_size(4 * sizeof(int))));

#define AS_GLOBAL __attribute__((address_space(1)))
#define AS_LDS    __attribute__((address_space(3)))

__device__ __forceinline__ int imin_i(int a, int b) { return a < b ? a : b; }

// One block per batch row. Dynamic LDS layout:
//   [0, SORT_M floats)                : neg-CE sort buffer (64 KB)
//   [SORT_M floats, + P float4)       : priors in cxcywh (140 KB)
__global__ void __launch_bounds__(TPB)
ssd_batch_kernel(const float* __restrict__ plocs,
                 const float* __restrict__ pscores,
                 const float* __restrict__ boxes,
                 const float* __restrict__ priors,
                 const int*   __restrict__ labels,
                 float* __restrict__ ws, int B, int P)
{
    const int b    = blockIdx.x;
    const int tid  = threadIdx.x;
    const int lane = tid & 31;

    extern __shared__ char smem[];
    float*  s_sort = (float*)smem;
    float4* s_pri  = (float4*)(smem + (size_t)SORT_M * sizeof(float));

    __shared__ float4             s_bx[NOBJ];   // corner boxes
    __shared__ float              s_ba[NOBJ];   // box areas
    __shared__ int                s_lab[NOBJ];
    __shared__ unsigned long long s_key[NOBJ];  // packed (iou_bits<<32)|(~p)
    __shared__ int                s_bestp[NOBJ];
    __shared__ int                s_numpos;
    __shared__ float              s_loc, s_pce, s_hard;

    if (tid < NOBJ) {
        float4 bx = ((const float4*)boxes)[(size_t)b * NOBJ + tid]; // xmin,ymin,xmax,ymax
        s_bx[tid]  = bx;
        s_ba[tid]  = (bx.z - bx.x) * (bx.w - bx.y);
        s_lab[tid] = labels[(size_t)b * NOBJ + tid];
        s_key[tid] = 0ull;
    }
    if (tid == 0) { s_numpos = 0; s_loc = 0.f; s_pce = 0.f; s_hard = 0.f; }

    // ---- stage priors into LDS (async CDNA5 path if available) ----
    const float4* pri4 = (const float4*)priors;
#ifdef USE_ASYNC_LDS
    for (int p = tid; p < P; p += TPB) {
        AS_GLOBAL v4i_b128* g = (AS_GLOBAL v4i_b128*)(pri4 + p);
        AS_LDS    v4i_b128* l = (AS_LDS    v4i_b128*)(s_pri + p);
        __builtin_amdgcn_global_load_async_to_lds_b128(g, l, 0, 0);
    }
    __builtin_amdgcn_s_wait_asynccnt(0);
#else
    for (int p = tid; p < P; p += TPB) s_pri[p] = pri4[p];
#endif
    __syncthreads();

    // ---- Phase A: per-object best prior (argmax over P, lowest index ties) ----
    for (int n = 0; n < NOBJ; ++n) {
        const float4 bx = s_bx[n];
        const float  ba = s_ba[n];
        unsigned long long best = 0ull;
        for (int p = tid; p < P; p += TPB) {
            float4 pr  = s_pri[p];
            float px0 = pr.x - pr.z * 0.5f, py0 = pr.y - pr.w * 0.5f;
            float px1 = pr.x + pr.z * 0.5f, py1 = pr.y + pr.w * 0.5f;
            float ix = fminf(bx.z, px1) - fmaxf(bx.x, px0);
            float iy = fminf(bx.w, py1) - fmaxf(bx.y, py0);
            ix = fmaxf(ix, 0.f); iy = fmaxf(iy, 0.f);
            float inter = ix * iy;
            float iou   = inter / (ba + pr.z * pr.w - inter);
            unsigned long long key =
                ((unsigned long long)__float_as_uint(iou) << 32) |
                (unsigned long long)(0xFFFFFFFFu - (unsigned)p);
            best = (key > best) ? key : best;
        }
        for (int off = 16; off > 0; off >>= 1) {
            unsigned long long o = __shfl_xor(best, off, 32);
            best = (o > best) ? o : best;
        }
        if (lane == 0) atomicMax(&s_key[n], best);
    }
    __syncthreads();
    if (tid < NOBJ)
        s_bestp[tid] = (int)(0xFFFFFFFFu - (unsigned)(s_key[tid] & 0xFFFFFFFFull));
    __syncthreads();

    // ---- Phase B: per-prior match, CE, loc-L1, neg-CE into sort buffer ----
    const float4* sc4 = (const float4*)(pscores + (size_t)b * P * NCLS);
    const float4* lc4 = (const float4*)(plocs   + (size_t)b * P * 4);

    for (int p = tid; p < P; p += TPB) {
        if (p + TPB < P) __builtin_prefetch(sc4 + p + TPB, 0, 1);
        float4 pr  = s_pri[p];
        float px0 = pr.x - pr.z * 0.5f, py0 = pr.y - pr.w * 0.5f;
        float px1 = pr.x + pr.z * 0.5f, py1 = pr.y + pr.w * 0.5f;
        float pa  = pr.z * pr.w;

        float best_ov = -1.f; int obj = 0;
        #pragma unroll 4
        for (int n = 0; n < NOBJ; ++n) {
            float4 bx = s_bx[n];
            float ix = fminf(bx.z, px1) - fmaxf(bx.x, px0);
            float iy = fminf(bx.w, py1) - fmaxf(bx.y, py0);
            ix = fmaxf(ix, 0.f); iy = fmaxf(iy, 0.f);
            float inter = ix * iy;
            float iou   = inter / (s_ba[n] + pa - inter);
            if (iou > best_ov) { best_ov = iou; obj = n; }  // first max wins
        }
        // forced matches: last object claiming this prior wins (scatter semantics)
        bool forced = false;
        for (int n = 0; n < NOBJ; ++n)
            if (s_bestp[n] == p) { obj = n; forced = true; }
        if (forced) best_ov = 1.0f;

        int lab = s_lab[obj];
        if (best_ov < THRESH) lab = 0;

        float4 sc = sc4[p];
        float m  = fmaxf(fmaxf(sc.x, sc.y), fmaxf(sc.z, sc.w));
        float se = __expf(sc.x - m) + __expf(sc.y - m) +
                   __expf(sc.z - m) + __expf(sc.w - m);
        float lse = m + __logf(se);
        float xl  = (lab == 0) ? sc.x : (lab == 1 ? sc.y : (lab == 2 ? sc.z : sc.w));
        float ce  = lse - xl;

        float nce = ce;
        if (lab != 0) {
            nce = 0.f;
            atomicAdd(&s_numpos, 1);
            atomicAdd(&s_pce, ce);
            float4 bb  = s_bx[obj];
            float bcx = 0.5f * (bb.x + bb.z), bcy = 0.5f * (bb.y + bb.w);
            float bw  = bb.z - bb.x,          bh  = bb.w - bb.y;
            float g0 = (bcx - pr.x) * 10.f / pr.z;
            float g1 = (bcy - pr.y) * 10.f / pr.w;
            float g2 = __logf(bw / pr.z) * 5.f;
            float g3 = __logf(bh / pr.w) * 5.f;
            float4 pl = lc4[p];
            float l1 = fabsf(pl.x - g0) + fabsf(pl.y - g1) +
                       fabsf(pl.z - g2) + fabsf(pl.w - g3);
            atomicAdd(&s_loc, l1);
        }
        s_sort[p] = nce;
    }
    for (int i = P + tid; i < SORT_M; i += TPB) s_sort[i] = -1.0f; // ce >= 0 pad
    __syncthreads();

    // ---- descending bitonic sort of neg-CE in LDS ----
    for (int kk = 2; kk <= SORT_M; kk <<= 1) {
        for (int j = kk >> 1; j > 0; j >>= 1) {
            for (int i = tid; i < SORT_M; i += TPB) {
                int ixj = i ^ j;
                if (ixj > i) {
                    float a = s_sort[i], c = s_sort[ixj];
                    bool up = ((i & kk) == 0);
                    if (up ? (a < c) : (a > c)) { s_sort[i] = c; s_sort[ixj] = a; }
                }
            }
            __syncthreads();
        }
    }

    // ---- top-k (k = 3*num_pos) hard-negative sum ----
    int kh = imin_i(3 * s_numpos, P);
    float hs = 0.f;
    for (int i = tid; i < kh; i += TPB) hs += s_sort[i];
    for (int off = 16; off > 0; off >>= 1) hs += __shfl_xor(hs, off, 32);
    if (lane == 0) atomicAdd(&s_hard, hs);
    __syncthreads();

    if (tid == 0) {
        ws[0 * B + b] = s_loc;
        ws[1 * B + b] = s_pce;
        ws[2 * B + b] = s_hard;
        ws[3 * B + b] = (float)s_numpos;
    }
}

// Single wave32. Stages the 4 partial arrays into zero-padded LDS, then reduces
// them with V_WMMA_F32_16X16X4_F32 (ones B-matrix => row-sum accumulation),
// keeping EXEC all-ones around the WMMAs (no guarded loads).
__global__ void __launch_bounds__(32)
ssd_final_kernel(const float* __restrict__ ws, float* __restrict__ out, int B)
{
    const int lane = threadIdx.x;
    __shared__ float s_arr[4 * BMAX];

    for (int i = lane; i < 4 * BMAX; i += 32) s_arr[i] = 0.0f;
    __syncthreads();
    const int Bc = imin_i(B, BMAX);
    for (int a = 0; a < 4; ++a)
        for (int i = lane; i < Bc; i += 32)
            s_arr[a * BMAX + i] = ws[(size_t)a * B + i];
    __syncthreads();

    float sums[4];
#ifdef USE_WMMA_RED
    const int ntile = (Bc + 63) / 64;
    v2f bones; bones.x = 1.0f; bones.y = 1.0f;
    const int m  = lane & 15;
    const int kb = (lane >> 4) << 1;
    for (int a = 0; a < 4; ++a) {
        v8f c = {};
        for (int t = 0; t < ntile; ++t) {
            int i0 = a * BMAX + t * 64 + m * 4 + kb;  // A is 16x4 row-major per tile
            v2f av;
            av.x = s_arr[i0];
            av.y = s_arr[i0 + 1];
            c = __builtin_amdgcn_wmma_f32_16x16x4_f32(
                    false, av, false, bones, (short)0, c, false, false);
        }
        float s = c[0] + c[1] + c[2] + c[3] + c[4] + c[5] + c[6] + c[7];
        for (int off = 16; off > 0; off >>= 1) s += __shfl_xor(s, off, 32);
        sums[a] = s * (1.0f / 16.0f);   // every rowsum replicated across 16 cols
    }
#else
    for (int a = 0; a < 4; ++a) {
        float s = 0.f;
        for (int i = 0; i < 4; ++i) { }  // keep structure minimal
        for (int i = lane; i < Bc; i += 32) s += s_arr[a * BMAX + i];
        for (int off = 16; off > 0; off >>= 1) s += __shfl_xor(s, off, 32);
        sums[a] = s;
    }
#endif
    if (lane == 0) {
        float npos      = sums[3];
        float loc_loss  = sums[0] / (npos * 4.0f);
        float conf_loss = (sums[2] + sums[1]) / npos;
        out[0] = conf_loss + loc_loss;
    }
}

extern "C" void kernel_launch(void* const* d_in, const int* in_sizes, int n_in,
                              void* d_out, int out_size, void* d_ws, size_t ws_size,
                              hipStream_t stream)
{
    const float* plocs   = (const float*)d_in[0];
    const float* pscores = (const float*)d_in[1];
    const float* boxes   = (const float*)d_in[2];
    const float* priors  = (const float*)d_in[3];
    const int*   labels  = (const int*)d_in[4];

    const int P = in_sizes[3] / 4;      // 8732
    const int B = in_sizes[4] / NOBJ;   // 128
    float* ws = (float*)d_ws;           // 4 arrays of B floats

    size_t lds = (size_t)SORT_M * sizeof(float) + (size_t)P * sizeof(float4); // ~204 KB of 320 KB/WGP
    ssd_batch_kernel<<<B, TPB, lds, stream>>>(plocs, pscores, boxes, priors, labels, ws, B, P);
    ssd_final_kernel<<<1, 32, 0, stream>>>(ws, (float*)d_out, B);
}